// GPTSanJapaneseTop1Router_343597384008
// MI455X (gfx1250) — compile-verified
//
#include <hip/hip_runtime.h>
#include <hip/hip_bf16.h>
#include <stdint.h>

#define NUM_EXPERTS 8
#define EXPERT_CAPACITY 320
#define G_DIM 8
#define T_DIM 2048
#define H_DIM 2048
#define ROWS (G_DIM * T_DIM)   // 16384 tokens total

typedef __attribute__((ext_vector_type(2))) float v2f;
typedef __attribute__((ext_vector_type(8))) float v8f;

// ---------------------------------------------------------------------------
// Kernel 1: router logits GEMM via V_WMMA_F32_16X16X4_F32.
//   A tile: 16 tokens x 4 hidden  (fp32, per-lane float2 global load)
//   B tile: 4 hidden x 16 "experts" (8 real + 8 zero columns, from LDS)
//   C tile: 16x16 fp32, columns 0..7 are the logits.
// W (8x2048 fp32 = 64KB) + one zero row staged in LDS once per block.
// A stream is software-pipelined with a 2-deep register double buffer so
// 8-16 global_load_b64 stay outstanding across iterations (no latency
// bubble at the head of each 32-k stage) -- this kernel is HBM-bound.
// ---------------------------------------------------------------------------
__global__ __launch_bounds__(256) void router_gemm_kernel(
    const float* __restrict__ hidden,   // [ROWS, H]
    const float* __restrict__ W,        // [E, H]
    const float* __restrict__ bias,     // [E]
    float* __restrict__ logits)         // [ROWS, E]
{
    __shared__ float sW[(NUM_EXPERTS + 1) * H_DIM];   // row 8 = zeros (72KB)

    const int tid = threadIdx.x;

    // Cooperative staging of W (+ zero pad row) into LDS, float4 vectorized.
    {
        const float4* W4  = (const float4*)W;
        float4*       sW4 = (float4*)sW;
        const int nW4 = NUM_EXPERTS * H_DIM / 4;   // 4096
        for (int i = tid; i < nW4; i += 256) sW4[i] = W4[i];
        const float4 z4 = make_float4(0.f, 0.f, 0.f, 0.f);
        const int nZ4 = H_DIM / 4;                 // 512
        for (int i = tid; i < nZ4; i += 256) sW4[nW4 + i] = z4;
    }
    __syncthreads();

    const int lane  = tid & 31;
    const int wave  = tid >> 5;            // 8 waves -> 128 tokens per block
    const int m     = lane & 15;           // M (token row) for A; N (expert col) for B/C
    const int khalf = (lane >> 4) << 1;    // 0 for lanes 0-15, 2 for lanes 16-31
    const int row0  = (blockIdx.x * 8 + wave) * 16;

    // A fragment source: lane holds {A[m][k+khalf], A[m][k+khalf+1]}
    const float* aBase = hidden + (size_t)(row0 + m) * H_DIM + khalf;
    // B fragment source: lane holds {W[n][k+khalf], W[n][k+khalf+1]}; n>=8 -> zero row
    const int nclamped = (m < NUM_EXPERTS) ? m : NUM_EXPERTS;
    const int bBase    = nclamped * H_DIM + khalf;

    // --- software pipeline: 32-k stages, prefetch distance 2 stages ---
    v2f abuf[2][8];
#pragma unroll
    for (int j = 0; j < 8; ++j) abuf[0][j] = *(const v2f*)(aBase + j * 4);
#pragma unroll
    for (int j = 0; j < 8; ++j) abuf[1][j] = *(const v2f*)(aBase + 32 + j * 4);

    v8f c = {};
#pragma unroll 2
    for (int k = 0; k < H_DIM; k += 32) {
        const int cur = (k >> 5) & 1;

        // Consume the stage loaded two iterations ago.
#pragma unroll
        for (int j = 0; j < 8; ++j) {
            v2f bf = *(const v2f*)&sW[bBase + k + j * 4];
            // (neg_a, A, neg_b, B, c_mod, C, reuse_a, reuse_b)
            c = __builtin_amdgcn_wmma_f32_16x16x4_f32(false, abuf[cur][j], false, bf,
                                                      (short)0, c, false, false);
        }

        // Prefetch stage k+64 into the slot just freed (wraps in-bounds on tail).
        const int kp = (k + 64) & (H_DIM - 1);
#pragma unroll
        for (int j = 0; j < 8; ++j)
            abuf[cur][j] = *(const v2f*)(aBase + kp + j * 4);
    }

    // C/D layout: lane holds column N = lane%16, VGPR r holds row M = r + 8*(lane/16)
    if (m < NUM_EXPERTS) {
        const float bn    = bias[m];
        const int   mbase = (lane >> 4) << 3;
#pragma unroll
        for (int r = 0; r < 8; ++r) {
            const int row = row0 + mbase + r;
            logits[(size_t)row * NUM_EXPERTS + m] = c[r] + bn;
        }
    }
}

// ---------------------------------------------------------------------------
// Kernel 2: per-token softmax max-prob + first-occurrence argmax.
// max router prob = 1 / sum_e exp(l_e - l_max).
// ---------------------------------------------------------------------------
__global__ __launch_bounds__(256) void router_softmax_kernel(
    const float* __restrict__ logits,     // [ROWS, E]
    float* __restrict__ probs_max,        // [ROWS]
    int* __restrict__ choice)             // [ROWS] scratch
{
    const int t = blockIdx.x * 256 + threadIdx.x;
    if (t >= ROWS) return;

    const float* lrow = logits + (size_t)t * NUM_EXPERTS;
    float l[NUM_EXPERTS];
    const float4 lo = *(const float4*)(lrow);
    const float4 hi = *(const float4*)(lrow + 4);
    l[0] = lo.x; l[1] = lo.y; l[2] = lo.z; l[3] = lo.w;
    l[4] = hi.x; l[5] = hi.y; l[6] = hi.z; l[7] = hi.w;

    float mx = l[0];
    int   am = 0;
#pragma unroll
    for (int e = 1; e < NUM_EXPERTS; ++e) {
        if (l[e] > mx) { mx = l[e]; am = e; }   // strict '>' keeps first occurrence
    }
    float s = 0.f;
#pragma unroll
    for (int e = 0; e < NUM_EXPERTS; ++e) s += __expf(l[e] - mx);

    probs_max[t] = 1.0f / s;
    choice[t]    = am;
}

// ---------------------------------------------------------------------------
// Kernel 3: per-group capacity cumsum + gated one-hot expert_index.
// One block per group; thread t owns 8 consecutive tokens.
// ---------------------------------------------------------------------------
__global__ __launch_bounds__(256) void router_capacity_kernel(
    const int* __restrict__ choice,       // [ROWS]
    int* __restrict__ expert_index)       // [ROWS, E] (int32 bits)
{
    __shared__ int sCnt[256 * NUM_EXPERTS];   // 8KB

    const int g        = blockIdx.x;
    const int tid      = threadIdx.x;
    const int tok_per  = T_DIM / 256;         // 8
    const int tok0     = g * T_DIM + tid * tok_per;

    int local[NUM_EXPERTS];
#pragma unroll
    for (int e = 0; e < NUM_EXPERTS; ++e) local[e] = 0;
    int ch[8];
#pragma unroll
    for (int i = 0; i < 8; ++i) { ch[i] = choice[tok0 + i]; local[ch[i]]++; }
#pragma unroll
    for (int e = 0; e < NUM_EXPERTS; ++e) sCnt[tid * NUM_EXPERTS + e] = local[e];
    __syncthreads();

    // Exclusive scan over 256 chunks, one expert column per lane (tiny, serial).
    if (tid < NUM_EXPERTS) {
        int run = 0;
        for (int i = 0; i < 256; ++i) {
            const int idx = i * NUM_EXPERTS + tid;
            const int tmp = sCnt[idx];
            sCnt[idx] = run;
            run += tmp;
        }
    }
    __syncthreads();

    int run[NUM_EXPERTS];
#pragma unroll
    for (int e = 0; e < NUM_EXPERTS; ++e) run[e] = sCnt[tid * NUM_EXPERTS + e];

#pragma unroll
    for (int i = 0; i < 8; ++i) {
        const int tok = tok0 + i;
        const int e   = ch[i];
        run[e] += 1;                                   // inclusive priority
        const int keep = (run[e] <= EXPERT_CAPACITY) ? 1 : 0;
        int* orow = expert_index + (size_t)tok * NUM_EXPERTS;
#pragma unroll
        for (int j = 0; j < NUM_EXPERTS; ++j) orow[j] = (j == e) ? keep : 0;
    }
}

// ---------------------------------------------------------------------------
extern "C" void kernel_launch(void* const* d_in, const int* in_sizes, int n_in,
                              void* d_out, int out_size, void* d_ws, size_t ws_size,
                              hipStream_t stream)
{
    const float* hidden = (const float*)d_in[0];   // [G,T,H] fp32
    const float* W      = (const float*)d_in[1];   // [E,H]   fp32
    const float* bias   = (const float*)d_in[2];   // [E]     fp32

    float* out          = (float*)d_out;
    int*   expert_index = (int*)out;                                  // [ROWS*E]
    float* probs_max    = out + (size_t)ROWS * NUM_EXPERTS;           // [ROWS]
    float* logits       = out + (size_t)ROWS * NUM_EXPERTS + ROWS;    // [ROWS*E]

    int* choice = (int*)d_ws;                                         // [ROWS]

    router_gemm_kernel<<<ROWS / 128, 256, 0, stream>>>(hidden, W, bias, logits);
    router_softmax_kernel<<<ROWS / 256, 256, 0, stream>>>(logits, probs_max, choice);
    router_capacity_kernel<<<G_DIM, 256, 0, stream>>>(choice, expert_index);
}